// MHSA_42545946034970
// MI455X (gfx1250) — compile-verified
//
#include <hip/hip_runtime.h>
#include <stdint.h>

// Reference: B=512, N=2048, F=16, amp=0.2, thr=0.5
#define N_COLS   2048
#define F_LEN    16
#define ROWB_OFF 2052   // dwords: row-B LDS offset = 8208B (16B aligned, +4 bank skew)

__device__ __forceinline__ void async_wait0()  { asm volatile("s_wait_asynccnt 0"  ::: "memory"); }
__device__ __forceinline__ void async_wait16() { asm volatile("s_wait_asynccnt 16" ::: "memory"); }

// CDNA5 async DMA: 32 lanes x 16B = 512B = 128 floats per instruction (GVS addressing).
__device__ __forceinline__ void async_b128(uint32_t lds_off, uint64_t sbase, uint32_t voff) {
  asm volatile("global_load_async_to_lds_b128 %0, %1, %2"
               :: "v"(lds_off), "v"(voff), "s"(sbase) : "memory");
}

// DPP16 move within 16-lane rows; invalid source lanes -> 0.
template <int CTRL>
__device__ __forceinline__ float dpp_mov(float v) {
  return __int_as_float(__builtin_amdgcn_update_dpp(0, __float_as_int(v), CTRL, 0xf, 0xf, true));
}
#define DPP_ROW_SHR(n) (0x110 | (n))
#define DPP_ROW_SHL(n) (0x100 | (n))

__device__ __forceinline__ unsigned ballot32(bool p) {
#if __has_builtin(__builtin_amdgcn_ballot_w32)
  return __builtin_amdgcn_ballot_w32(p);
#else
  return (unsigned)__ballot(p);
#endif
}

__global__ __launch_bounds__(32)
void mhsa_encode(const float* __restrict__ data,
                 const float* __restrict__ filt,
                 float* __restrict__ out) {
  __shared__ float lbuf[ROWB_OFF + N_COLS];   // row A at 0, row B at ROWB_OFF

  const int  lane = threadIdx.x;      // 0..31
  const int  j    = lane & 15;        // filter tap owned by this lane
  const bool hi   = lane >= 16;       // lanes 16..31 -> row B
  const int  rowA = blockIdx.x * 2;

  // ---- stage both rows into LDS via async DMA (two phases) ----
  const uint64_t gbase = (uint64_t)(uintptr_t)data + (uint64_t)rowA * (N_COLS * 4ull);
  const uint32_t l0   = (uint32_t)(uintptr_t)&lbuf[0];
  const uint32_t lB   = l0 + ROWB_OFF * 4;
  const uint32_t lb16 = (uint32_t)lane * 16u;
#pragma unroll 1
  for (int c0 = 0; c0 < 1024; c0 += 128) {            // phase 0: 16 instrs, cols 0..1023
    async_b128(l0 + c0 * 4 + lb16, gbase, (uint32_t)(c0 * 4) + lb16);
    async_b128(lB + c0 * 4 + lb16, gbase, (uint32_t)(N_COLS * 4 + c0 * 4) + lb16);
  }
#pragma unroll 1
  for (int c0 = 1024; c0 < 2048; c0 += 128) {         // phase 1: 16 instrs, cols 1024..2047
    async_b128(l0 + c0 * 4 + lb16, gbase, (uint32_t)(c0 * 4) + lb16);
    async_b128(lB + c0 * 4 + lb16, gbase, (uint32_t)(N_COLS * 4 + c0 * 4) + lb16);
  }

  // per-lane filter constants: f[j] = filt[j]*0.2 ; g[j] = (j+1)*f[j]
  const float f = filt[j] * 0.2f;
  const float g = (float)(j + 1) * f;

  async_wait16();                                     // phase 0 resident

  const float* __restrict__ lrow = &lbuf[hi ? ROWB_OFF : 0];
  float w = lrow[j];                                  // window = row[0..15]
  float* __restrict__ orow = out + (size_t)(rowA + (hi ? 1 : 0)) * N_COLS;
  const bool writer = (j == 0);                       // lanes 0 / 16 emit outputs
  const bool last_t = (j == 15);

  // One i-step. k = #{prefix sums of max(0, g-w) <= 0.5}; out = (k>0); w -= k*f.
  auto step = [&]() -> float {
    float c = g - w;
    c = (c > 0.f) ? c : 0.f;
    float S = c;                                      // inclusive 16-lane DPP scan
    S += dpp_mov<DPP_ROW_SHR(1)>(S);
    S += dpp_mov<DPP_ROW_SHR(2)>(S);
    S += dpp_mov<DPP_ROW_SHR(4)>(S);
    S += dpp_mov<DPP_ROW_SHR(8)>(S);
    unsigned m = ballot32(S <= 0.5f);
    int kA = __popc(m & 0xFFFFu);
    int kB = __popc(m >> 16);
    float kf = hi ? (float)kB : (float)kA;
    w = fmaf(-kf, f, w);
    return fminf(kf, 1.0f);                           // kf integer-valued -> (k>0)?1:0
  };
  auto shift_in = [&](float nextv) {                  // w[j] = w[j+1]; w[15] = next element
    float up = dpp_mov<DPP_ROW_SHL(1)>(w);
    w = last_t ? nextv : up;
  };

  auto run_block = [&](int i0, int i1) {              // i0,i1 multiples of 4
#pragma unroll 1
    for (int i = i0; i < i1; i += 4) {
      float o0 = step(); shift_in(lrow[i + 16]);
      float o1 = step(); shift_in(lrow[i + 17]);
      float o2 = step(); shift_in(lrow[i + 18]);
      float o3 = step(); shift_in(lrow[i + 19]);
      if (writer) *(float4*)(orow + i) = make_float4(o0, o1, o2, o3);
    }
  };

  run_block(0, 1008);                                 // reads cols <= 1023 (phase 0)
  async_wait0();                                      // phase 1 resident
  run_block(1008, 2032);                              // reads cols <= 2047

  // i = 2032: last fitting step (no shift); i = 2033..2047: err stays 0 -> out = 1
  float oL = step();
  if (writer) {
    *(float4*)(orow + 2032) = make_float4(oL, 1.f, 1.f, 1.f);
    *(float4*)(orow + 2036) = make_float4(1.f, 1.f, 1.f, 1.f);
    *(float4*)(orow + 2040) = make_float4(1.f, 1.f, 1.f, 1.f);
    *(float4*)(orow + 2044) = make_float4(1.f, 1.f, 1.f, 1.f);
  }
}

extern "C" void kernel_launch(void* const* d_in, const int* in_sizes, int n_in,
                              void* d_out, int out_size, void* d_ws, size_t ws_size,
                              hipStream_t stream) {
  const float* data = (const float*)d_in[0];   // [B, 2048] f32
  const float* filt = (const float*)d_in[1];   // [16] f32
  float* out = (float*)d_out;                  // [B, 2048] f32

  const int B = in_sizes[0] / N_COLS;          // 512 rows
  dim3 grid(B / 2);                            // 2 rows per wave32 workgroup -> 256 blocks
  dim3 block(32);
  mhsa_encode<<<grid, block, 0, stream>>>(data, filt, out);
}